// pool_50165218017330
// MI455X (gfx1250) — compile-verified
//
#include <hip/hip_runtime.h>
#include <hip/hip_bf16.h>

// Cumulative mean over dim0 of x[8192][4096] fp32.
// out[s][c] = cumsum(x,0)[s][c] / (s+1)
//
// 3-pass column scan, chunk = 16 rows (512 chunks):
//   P1: ws[chunk][c] = sum of 16 rows of chunk            (coalesced)
//   P2: in-place exclusive prefix over chunks per column  (coalesced)
//   P3: per 16x16 tile: Y = L * X via 4x V_WMMA_F32_16X16X4_F32 (fp32 exact
//       0/1 products), add chunk offset, multiply by rcp(row+1), NT-store.

static constexpr int S      = 8192;
static constexpr int C      = 4096;
static constexpr int CHUNK  = 16;
static constexpr int NCHUNK = S / CHUNK;   // 512

typedef __attribute__((ext_vector_type(2))) float v2f;
typedef __attribute__((ext_vector_type(8))) float v8f;

// ---------------- Pass 1: per-chunk column sums ----------------
__global__ void cm_chunk_sums(const float* __restrict__ x,
                              float* __restrict__ ws) {
    const int col   = blockIdx.x * 256 + threadIdx.x;   // gridDim.x = C/256
    const int chunk = blockIdx.y;                       // gridDim.y = NCHUNK
    const float* p  = x + (size_t)chunk * CHUNK * C + col;
    float s = 0.0f;
#pragma unroll
    for (int r = 0; r < CHUNK; ++r) s += p[(size_t)r * C];
    ws[(size_t)chunk * C + col] = s;
}

// ---------------- Pass 2: exclusive prefix over chunks (in place) ----------
__global__ void cm_scan_chunks(float* __restrict__ ws) {
    const int col = blockIdx.x * 256 + threadIdx.x;     // gridDim.x = C/256
    float run = 0.0f;
#pragma unroll 8
    for (int k = 0; k < NCHUNK; ++k) {
        float t = ws[(size_t)k * C + col];
        ws[(size_t)k * C + col] = run;
        run += t;
    }
}

// ---------------- Pass 3: WMMA triangular tile scan ----------------
// One wave32 computes one 16x16 tile: Y = L(16x16 lower-tri ones) * X(16x16)
// as four chained V_WMMA_F32_16X16X4_F32 (K = 16 in steps of 4).
__global__ void cm_tile_scan(const float* __restrict__ x,
                             const float* __restrict__ off,
                             float* __restrict__ out) {
    const int lane = threadIdx.x & 31;
    const int wave = threadIdx.x >> 5;                  // 8 waves / block
    const int n    = lane & 15;                         // tile column (N), also M for A
    const int hi   = lane >> 4;                         // lane half

    const int c0    = blockIdx.x * 128 + wave * 16;     // gridDim.x = C/128
    const int chunk = blockIdx.y;                       // gridDim.y = NCHUNK
    const int r0    = chunk * CHUNK;

    v8f acc = {};                                       // C matrix = 0
#pragma unroll
    for (int k = 0; k < 4; ++k) {
        // A slice: columns [4k..4k+3] of L. ISA f32 16x4 layout:
        //   lane M = lane%16 ; VGPR v holds K_local = v + 2*hi
        // A[M][K] = (4k + K_local) <= M ? 1 : 0
        v2f a;
        a.x = ((4 * k + 0 + 2 * hi) <= n) ? 1.0f : 0.0f;
        a.y = ((4 * k + 1 + 2 * hi) <= n) ? 1.0f : 0.0f;

        // B slice: X rows [r0+4k .. r0+4k+3], cols [c0..c0+15].
        //   lane N = lane%16 ; VGPR v holds K_local = v + 2*hi
        const float* bp = x + (size_t)(r0 + 4 * k + 2 * hi) * C + (c0 + n);
        v2f b;
        b.x = bp[0];
        b.y = bp[C];

        acc = __builtin_amdgcn_wmma_f32_16x16x4_f32(
            /*neg_a=*/false, a, /*neg_b=*/false, b,
            /*c_mod=*/(short)0, acc, /*reuse_a=*/false, /*reuse_b=*/false);
    }

    // Exclusive prefix of all previous chunks for this column.
    const float carry = off[(size_t)chunk * C + (c0 + n)];

    // D layout: VGPR v -> row M = v + 8*hi, col N = n.
    // Scale by 1/(row+1) using v_rcp_f32 (exact-int divisor, <=1 ulp) and
    // store non-temporally so the 128MB output stream does not evict x from
    // the 192MB L2 (pass 1 already pulled x through L2 for reuse here).
    float* op = out + (size_t)(r0 + 8 * hi) * C + (c0 + n);
#pragma unroll
    for (int v = 0; v < 8; ++v) {
        const int row = r0 + 8 * hi + v;
        const float inv = __builtin_amdgcn_rcpf((float)(row + 1));
        __builtin_nontemporal_store((acc[v] + carry) * inv, op + (size_t)v * C);
    }
}

extern "C" void kernel_launch(void* const* d_in, const int* in_sizes, int n_in,
                              void* d_out, int out_size, void* d_ws, size_t ws_size,
                              hipStream_t stream) {
    const float* x = (const float*)d_in[0];
    float* out = (float*)d_out;
    float* ws  = (float*)d_ws;   // needs NCHUNK*C*4 = 8 MB

    // Pass 1: chunk column sums
    cm_chunk_sums<<<dim3(C / 256, NCHUNK), dim3(256), 0, stream>>>(x, ws);
    // Pass 2: exclusive prefix over chunk sums (in place, one thread per column)
    cm_scan_chunks<<<dim3(C / 256), dim3(256), 0, stream>>>(ws);
    // Pass 3: WMMA intra-chunk scan + carry + scale
    cm_tile_scan<<<dim3(C / 128, NCHUNK), dim3(256), 0, stream>>>(x, ws, out);
}